// Attention_67413806678604
// MI455X (gfx1250) — compile-verified
//
#include <hip/hip_runtime.h>
#include <math.h>

// Problem constants
#define B_    32
#define DIM_  4096
#define H_    32
#define KV_   8
#define HD_   128
#define G_    4      // H/KV
#define PREV_ 2048
#define STOT_ 2049   // PREV + 1
#define TS_   32     // V tile rows staged in LDS per step

typedef __attribute__((ext_vector_type(2))) float v2f;
typedef __attribute__((ext_vector_type(8))) float v8f;
typedef int i32x4_ __attribute__((vector_size(16)));

// ---------------------------------------------------------------------------
// Async global -> LDS staging helpers (gfx1250 ASYNCcnt path), with a
// synchronous fallback if the builtin is not declared by this toolchain.
// Builtin signature (from hipcc diagnostic): pointers to 16-byte int vectors.
// ---------------------------------------------------------------------------
__device__ __forceinline__ void stage_b128(const float* g, float* l) {
#if __has_builtin(__builtin_amdgcn_global_load_async_to_lds_b128)
  __builtin_amdgcn_global_load_async_to_lds_b128(
      (i32x4_*)(void*)g, (i32x4_*)(void*)l, 0, 0);
#else
  *(float4*)l = *(const float4*)g;
#endif
}

__device__ __forceinline__ void stage_wait_all() {
#if __has_builtin(__builtin_amdgcn_s_wait_asynccnt)
  __builtin_amdgcn_s_wait_asynccnt(0);
#endif
}

// ---------------------------------------------------------------------------
// GEMM: Y[32,N] = X[32,K] @ W[K,N], fp32, via V_WMMA_F32_16X16X4_F32.
// K, N are compile-time so the entire W walk folds into immediate offsets.
// One wave owns a 32x32 output tile: 4 accumulators (2 M-tiles x 2 N-tiles),
// A shared across the N-tiles, B shared across the M-tiles.
// A 16x4 layout: lane l -> row m=l&15; VGPR pair holds K=kh,kh+1, kh=2*(l>>4).
// B 4x16 layout: lane l -> col (l&15);  VGPR pair holds rows K=kh,kh+1.
// C/D: VGPR i -> row (l>=16 ? 8:0)+i, col l&15.
// ---------------------------------------------------------------------------
template <int K, int N>
__global__ void __launch_bounds__(128)
gemm32_wmma(const float* __restrict__ X, const float* __restrict__ W,
            float* __restrict__ Y) {
  const int lane = (int)(threadIdx.x & 31);
  const int wave = (int)(threadIdx.x >> 5);
  const int n0   = ((int)blockIdx.x * 4 + wave) * 32;  // 32-wide n tile
  const int mr   = lane & 15;
  const int kh   = (lane >> 4) << 1;                   // 0 or 2
  const int nA   = n0 + mr;
  const int nB   = nA + 16;

  const float* xr0 = X + mr * K + kh;          // row mr, phase kh
  const float* xr1 = xr0 + 16 * K;             // row mr+16
  const float* wp  = W + kh * N + nA;          // row phase kh, col nA

  v8f c00 = {}, c01 = {}, c10 = {}, c11 = {};

  for (int kb = 0; kb < K; kb += 32) {
    // Prefetch W ~2 k-blocks ahead into the cache hierarchy.
    __builtin_prefetch(wp + (kb + 64) * N, 0, 1);
#pragma unroll
    for (int ku = 0; ku < 32; ku += 4) {
      const int k0 = kb + ku;
      v2f a0, a1, b0, b1;
      a0.x = xr0[k0];
      a0.y = xr0[k0 + 1];
      a1.x = xr1[k0];
      a1.y = xr1[k0 + 1];
      const float* wr = wp + k0 * N;           // constant-strided offsets
      b0.x = wr[0];
      b0.y = wr[N];
      b1.x = wr[16];
      b1.y = wr[N + 16];
      c00 = __builtin_amdgcn_wmma_f32_16x16x4_f32(
          false, a0, false, b0, (short)0, c00, false, false);
      c10 = __builtin_amdgcn_wmma_f32_16x16x4_f32(
          false, a1, false, b0, (short)0, c10, false, false);
      c01 = __builtin_amdgcn_wmma_f32_16x16x4_f32(
          false, a0, false, b1, (short)0, c01, false, false);
      c11 = __builtin_amdgcn_wmma_f32_16x16x4_f32(
          false, a1, false, b1, (short)0, c11, false, false);
    }
  }

  const int cm = (lane >> 4) << 3;                     // 0 or 8
#pragma unroll
  for (int i = 0; i < 8; ++i) {
    Y[(cm + i) * N + nA]      = c00[i];
    Y[(cm + i + 16) * N + nA] = c10[i];
    Y[(cm + i) * N + nB]      = c01[i];
    Y[(cm + i + 16) * N + nB] = c11[i];
  }
}

// ---------------------------------------------------------------------------
// RoPE applied in-place to q [B,H,HD] and k_new [B,KV,HD].
// ---------------------------------------------------------------------------
__global__ void rope_kernel(float* __restrict__ q, float* __restrict__ knew,
                            const float* __restrict__ freqs) {
  const int QP = B_ * H_ * (HD_ / 2);
  const int KP = B_ * KV_ * (HD_ / 2);
  int idx = (int)(blockIdx.x * blockDim.x + threadIdx.x);
  if (idx >= QP + KP) return;

  float* t;
  int i;
  if (idx < QP) {
    t = q + (idx >> 6) * HD_;
    i = idx & 63;
  } else {
    int j = idx - QP;
    t = knew + (j >> 6) * HD_;
    i = j & 63;
  }
  const float f = freqs[i];
  const float c = cosf(f), s = sinf(f);
  const float a = t[2 * i], b = t[2 * i + 1];
  t[2 * i]     = a * c - b * s;
  t[2 * i + 1] = a * s + b * c;
}

// ---------------------------------------------------------------------------
// Attention: one block per (b, kv). 512 threads = 16 waves.
// Phase 1: wave-per-key-row QK^T (lane covers 4 dims, shuffle reduce).
// Phase 2: block softmax (scores in LDS).
// Phase 3: AV with V tiles double-buffered in LDS via async global->LDS
//          copies (each V byte crosses the WGP boundary exactly once).
// Dynamic LDS: 2 * TS_ * HD_ floats (32 KB) for the V tiles.
// ---------------------------------------------------------------------------
__global__ void __launch_bounds__(512)
attn_kernel(const float* __restrict__ q,       // [B,H,HD]
            const float* __restrict__ knew,    // [B,KV,HD]
            const float* __restrict__ vnew,    // [B,KV,HD]
            const float* __restrict__ cache_k, // [B,PREV,KV,HD]
            const float* __restrict__ cache_v, // [B,PREV,KV,HD]
            float* __restrict__ out) {         // [B,H,HD]
  const int b    = (int)blockIdx.x >> 3;
  const int kv   = (int)blockIdx.x & 7;
  const int tid  = (int)threadIdx.x;
  const int lane = tid & 31;
  const int wave = tid >> 5;

  __shared__ float sc[G_][STOT_];
  __shared__ float red[G_][128];
  __shared__ float mx[G_], sm[G_];
  extern __shared__ float vt[];                // 2 * TS_ * HD_ floats

  const float scale = 0.08838834764831845f;    // 1/sqrt(128)
  const int d = lane * 4;

  float4 qv[G_];
#pragma unroll
  for (int g = 0; g < G_; ++g)
    qv[g] = *(const float4*)(q + (b * H_ + kv * G_ + g) * HD_ + d);

  // ---- Phase 1: scores ----
  const int krow = (b * PREV_ * KV_ + kv) * HD_;       // base of (b, s=0, kv)
  for (int s = wave; s < STOT_; s += 16) {
    const float* kp = (s < PREV_)
        ? cache_k + krow + s * (KV_ * HD_) + d
        : knew + (b * KV_ + kv) * HD_ + d;
    const float4 kvv = *(const float4*)kp;

    float pr[G_];
#pragma unroll
    for (int g = 0; g < G_; ++g)
      pr[g] = qv[g].x * kvv.x + qv[g].y * kvv.y +
              qv[g].z * kvv.z + qv[g].w * kvv.w;

#pragma unroll
    for (int off = 16; off > 0; off >>= 1)
#pragma unroll
      for (int g = 0; g < G_; ++g)
        pr[g] += __shfl_xor(pr[g], off, 32);

    if (lane == 0) {
#pragma unroll
      for (int g = 0; g < G_; ++g) sc[g][s] = pr[g] * scale;
    }
  }
  __syncthreads();

  // ---- Phase 2: softmax ----
  const int g    = tid >> 7;   // 0..3
  const int t128 = tid & 127;  // 0..127

  float m = -INFINITY;
  for (int s = t128; s < STOT_; s += 128) m = fmaxf(m, sc[g][s]);
  red[g][t128] = m;
  __syncthreads();
  for (int off = 64; off > 0; off >>= 1) {
    if (t128 < off) red[g][t128] = fmaxf(red[g][t128], red[g][t128 + off]);
    __syncthreads();
  }
  if (t128 == 0) mx[g] = red[g][0];
  __syncthreads();

  const float mg = mx[g];
  float lsum = 0.f;
  for (int s = t128; s < STOT_; s += 128) {
    const float e = __expf(sc[g][s] - mg);
    sc[g][s] = e;
    lsum += e;
  }
  red[g][t128] = lsum;
  __syncthreads();
  for (int off = 64; off > 0; off >>= 1) {
    if (t128 < off) red[g][t128] += red[g][t128 + off];
    __syncthreads();
  }
  if (t128 == 0) sm[g] = red[g][0];
  __syncthreads();

  // ---- Phase 3: AV with double-buffered async V staging ----
  const int dd = t128;
  const float* vbase = cache_v + (b * PREV_ * KV_ + kv) * HD_;
  const int NT = PREV_ / TS_;                  // 64 tiles
  const int CHUNKS = TS_ * (HD_ / 4);          // 1024 b128 chunks per tile

  // Prologue: issue tile 0 into buffer 0.
#pragma unroll
  for (int c = tid; c < CHUNKS; c += 512) {
    const int r = c >> 5, col = (c & 31) * 4;
    stage_b128(vbase + r * (KV_ * HD_) + col, vt + r * HD_ + col);
  }

  float acc = 0.f;
  for (int t = 0; t < NT; ++t) {
    stage_wait_all();
    __syncthreads();                            // tile t fully in LDS

    if (t + 1 < NT) {                           // issue tile t+1
      float* nb = vt + ((t + 1) & 1) * (TS_ * HD_);
      const float* gsrc = vbase + (t + 1) * TS_ * (KV_ * HD_);
#pragma unroll
      for (int c = tid; c < CHUNKS; c += 512) {
        const int r = c >> 5, col = (c & 31) * 4;
        stage_b128(gsrc + r * (KV_ * HD_) + col, nb + r * HD_ + col);
      }
    }

    const float* vb = vt + (t & 1) * (TS_ * HD_);
    const int s0 = t * TS_;
#pragma unroll
    for (int j = 0; j < TS_; ++j)
      acc += sc[g][s0 + j] * vb[j * HD_ + dd];
    __syncthreads();                            // done reading before rewrite
  }
  acc += sc[g][PREV_] * vnew[(b * KV_ + kv) * HD_ + dd];

  out[(b * H_ + kv * G_ + g) * HD_ + dd] = acc / sm[g];
}

// ---------------------------------------------------------------------------
extern "C" void kernel_launch(void* const* d_in, const int* in_sizes, int n_in,
                              void* d_out, int out_size, void* d_ws,
                              size_t ws_size, hipStream_t stream) {
  (void)in_sizes; (void)n_in; (void)out_size; (void)ws_size;

  const float* x       = (const float*)d_in[0];
  const float* wq      = (const float*)d_in[1];
  const float* wk      = (const float*)d_in[2];
  const float* wv      = (const float*)d_in[3];
  const float* wo      = (const float*)d_in[4];
  const float* cache_k = (const float*)d_in[5];
  const float* cache_v = (const float*)d_in[6];
  const float* freqs   = (const float*)d_in[7];

  float* ws = (float*)d_ws;
  float* qb = ws;                               // B*H*HD   = 131072
  float* kb = ws + 131072;                      // B*KV*HD  =  32768
  float* vb = ws + 131072 + 32768;              //          =  32768
  float* ab = ws + 131072 + 65536;              // B*H*HD   = 131072

  // QKV projections (WMMA fp32). 32-col tiles, 4 waves (tiles) per block.
  gemm32_wmma<DIM_, H_ * HD_><<<32, 128, 0, stream>>>(x, wq, qb);   // N=4096
  gemm32_wmma<DIM_, KV_ * HD_><<<8, 128, 0, stream>>>(x, wk, kb);   // N=1024
  gemm32_wmma<DIM_, KV_ * HD_><<<8, 128, 0, stream>>>(x, wv, vb);   // N=1024

  // RoPE on q and k_new: (B*H + B*KV)*64 = 81920 pairs
  rope_kernel<<<320, 256, 0, stream>>>(qb, kb, freqs);

  // Attention: one block per (b, kv); 32 KB dynamic LDS for V tiles.
  attn_kernel<<<B_ * KV_, 512, 2 * TS_ * HD_ * sizeof(float), stream>>>(
      qb, kb, vb, cache_k, cache_v, ab);

  // Output projection into d_out
  gemm32_wmma<DIM_, DIM_><<<32, 128, 0, stream>>>(ab, wo, (float*)d_out);
}